// GammatoneFilterbank_45595372814481
// MI455X (gfx1250) — compile-verified
//
#include <hip/hip_runtime.h>

// Problem constants (match the reference).
constexpr int B_   = 16;      // batch
constexpr int C_   = 50;      // channels (filters)
constexpr int T_   = 40000;   // samples
constexpr int LB_  = 800;     // block length (multiple of 16; T_ = NB_*LB_)
constexpr int NB_  = T_ / LB_;        // 50 blocks
constexpr int NCH_ = B_ * C_;         // 800 chains

typedef __attribute__((ext_vector_type(2))) float v2f;
typedef __attribute__((ext_vector_type(8))) float v8f;

// ---------------------------------------------------------------------------
// Per-channel cascade coefficients: b[s][c][0..2], a[s][c][0..2] (a0 == 1).
struct Coef { float b0[4], b1[4], b2[4], a1[4], a2[4]; };

__device__ __forceinline__ void load_coefs(const float* __restrict__ bc,
                                           const float* __restrict__ ac,
                                           int c, Coef& k) {
#pragma unroll
  for (int s = 0; s < 4; ++s) {
    const float* bp = bc + (s * C_ + c) * 3;
    const float* ap = ac + (s * C_ + c) * 3;
    k.b0[s] = bp[0]; k.b1[s] = bp[1]; k.b2[s] = bp[2];
    k.a1[s] = ap[1]; k.a2[s] = ap[2];
  }
}

// One sample through the 4-stage cascade. p1[s]/p2[s] = y_s[t-1]/y_s[t-2].
// xp1/xp2 = x[t-1]/x[t-2]. Returns stage-4 output; updates all state.
__device__ __forceinline__ float cascade_step(const Coef& k, float x,
                                              float& xp1, float& xp2,
                                              float p1[4], float p2[4]) {
  float u = x, up1 = xp1, up2 = xp2;
  xp2 = xp1; xp1 = x;
#pragma unroll
  for (int s = 0; s < 4; ++s) {
    float v = fmaf(k.b0[s], u, fmaf(k.b1[s], up1, k.b2[s] * up2));
    float y = fmaf(-k.a1[s], p1[s], fmaf(-k.a2[s], p2[s], v));
    up1 = p1[s]; up2 = p2[s];        // next stage's input history (old values)
    p2[s] = p1[s]; p1[s] = y;        // this stage's output history
    u = y;
  }
  return u;
}

// ---------------------------------------------------------------------------
// Kernel 1: per-channel unit-initial-state responses.
// Thread (c, j): run cascade with x==0, state component j == 1.
// Records final-stage output h[c][j][t] (t < LB_) and end 8-state -> E[c][i][j].
// State component index: i = 2*s -> y_s[-1], i = 2*s+1 -> y_s[-2].
__global__ void gt_precompute_h(const float* __restrict__ bc,
                                const float* __restrict__ ac,
                                float* __restrict__ h,   // [C_][8][LB_]
                                float* __restrict__ E) { // [C_][8][8]
  int tid = blockIdx.x * blockDim.x + threadIdx.x;
  if (tid >= C_ * 8) return;
  int c = tid >> 3, j = tid & 7;
  Coef k; load_coefs(bc, ac, c, k);
  float p1[4] = {0, 0, 0, 0}, p2[4] = {0, 0, 0, 0};
  if (j & 1) p2[j >> 1] = 1.0f; else p1[j >> 1] = 1.0f;
  float xp1 = 0.f, xp2 = 0.f;
  float* hp = h + (c * 8 + j) * LB_;
  for (int t = 0; t < LB_; ++t)
    hp[t] = cascade_step(k, 0.0f, xp1, xp2, p1, p2);
#pragma unroll
  for (int s = 0; s < 4; ++s) {
    E[(c * 8 + 2 * s    ) * 8 + j] = p1[s];
    E[(c * 8 + 2 * s + 1) * 8 + j] = p2[s];
  }
}

// ---------------------------------------------------------------------------
// Kernel 2: zero-state pass. One thread per (block, chain); adjacent threads
// share the same batch's x reads (broadcast-friendly). Writes y0 into `out`
// (corrected later in place) and the end 8-state into dstates.
__global__ void gt_zero_state(const float* __restrict__ x,
                              const float* __restrict__ bc,
                              const float* __restrict__ ac,
                              float* __restrict__ out,       // y0: [B_][C_][T_]
                              float* __restrict__ dstates) { // [NCH_][NB_][8]
  int tid = blockIdx.x * blockDim.x + threadIdx.x;
  if (tid >= NCH_ * NB_) return;
  int kb = tid / NCH_;
  int chain = tid % NCH_;
  int b = chain / C_, c = chain % C_;
  Coef k; load_coefs(bc, ac, c, k);

  const float* xp = x + (size_t)b * T_ + kb * LB_;
  float xp1 = (kb > 0) ? xp[-1] : 0.0f;   // real x history across block edge
  float xp2 = (kb > 0) ? xp[-2] : 0.0f;
  float p1[4] = {0, 0, 0, 0}, p2[4] = {0, 0, 0, 0};

  float* op = out + (size_t)chain * T_ + kb * LB_;
  for (int t = 0; t < LB_; t += 4) {
    float4 xi = *(const float4*)(xp + t);
    float4 o;
    o.x = cascade_step(k, xi.x, xp1, xp2, p1, p2);
    o.y = cascade_step(k, xi.y, xp1, xp2, p1, p2);
    o.z = cascade_step(k, xi.z, xp1, xp2, p1, p2);
    o.w = cascade_step(k, xi.w, xp1, xp2, p1, p2);
    *(float4*)(op + t) = o;
  }
  float* dp = dstates + ((size_t)chain * NB_ + kb) * 8;
#pragma unroll
  for (int s = 0; s < 4; ++s) { dp[2 * s] = p1[s]; dp[2 * s + 1] = p2[s]; }
}

// ---------------------------------------------------------------------------
// Kernel 3: cross-block boundary scan per chain:  S_{k+1} = E_c * S_k + d_k.
// Stores the INCOMING state of each block into Sws.
__global__ void gt_scan(const float* __restrict__ E,
                        const float* __restrict__ dstates,
                        float* __restrict__ Sws) { // [NCH_][NB_][8]
  int chain = blockIdx.x * blockDim.x + threadIdx.x;
  if (chain >= NCH_) return;
  int c = chain % C_;
  float Ec[8][8];
#pragma unroll
  for (int i = 0; i < 8; ++i)
#pragma unroll
    for (int j = 0; j < 8; ++j) Ec[i][j] = E[(c * 8 + i) * 8 + j];

  float s[8] = {0, 0, 0, 0, 0, 0, 0, 0};
  for (int kb = 0; kb < NB_; ++kb) {
    size_t off = ((size_t)chain * NB_ + kb) * 8;
#pragma unroll
    for (int i = 0; i < 8; ++i) Sws[off + i] = s[i];
    const float* dp = dstates + off;
    float ns[8];
#pragma unroll
    for (int i = 0; i < 8; ++i) {
      float acc = dp[i];
#pragma unroll
      for (int j = 0; j < 8; ++j) acc = fmaf(Ec[i][j], s[j], acc);
      ns[i] = acc;
    }
#pragma unroll
    for (int i = 0; i < 8; ++i) s[i] = ns[i];
  }
}

// ---------------------------------------------------------------------------
// Kernel 4: WMMA fix-up (in place on `out`):
//   D(16 t x 16 batches) = H(16x8) * S(8x16) + y0tile, via two chained
//   V_WMMA_F32_16X16X4_F32 (K = 0..3 then 4..7).
// One wave per tile; blockDim=256 => 8 tiles/workgroup; grid sized exactly so
// EXEC is all ones (WMMA requirement). C/D layout per ISA: lanes 0-15 hold
// N=lane, M=0..7 in VGPR0..7; lanes 16-31 hold N=lane-16, M=8..15.
// A (16x4): lanes 0-15 -> M=lane, K=0/1; lanes 16-31 -> M=lane-16, K=2/3.
__global__ void gt_fixup_wmma(const float* __restrict__ h,   // [C_][8][LB_]
                              const float* __restrict__ Sws, // [NCH_][NB_][8]
                              float* __restrict__ out) {     // [B_][C_][T_]
  int wave = blockIdx.x * (blockDim.x >> 5) + (threadIdx.x >> 5);
  int lane = threadIdx.x & 31;
  int c  = wave / (T_ / 16);
  int tt = wave % (T_ / 16);
  int t0 = tt * 16;
  int kb = t0 / LB_;
  int half = lane >> 4;   // 0: rows M=0..7 / K=0,1   1: rows M=8..15 / K=2,3
  int l    = lane & 15;   // N (batch) for B/C/D;  M (t offset) for A

  // C tile: y0, 8 consecutive t per lane (N = batch l, M-range by half).
  float* yp = out + ((size_t)(l * C_ + c)) * T_ + t0 + half * 8;
  float4 c0 = *(const float4*)(yp);
  float4 c1 = *(const float4*)(yp + 4);
  v8f acc;
  acc[0] = c0.x; acc[1] = c0.y; acc[2] = c0.z; acc[3] = c0.w;
  acc[4] = c1.x; acc[5] = c1.y; acc[6] = c1.z; acc[7] = c1.w;

  int k0 = half * 2;                               // K index base per half
  const float* hp = h + (size_t)(c * 8) * LB_ + t0 + l;  // h[c][j][t0+l] at j*LB_
  v2f a_lo = { hp[(k0    ) * LB_], hp[(k0 + 1) * LB_] }; // K = 0..3
  v2f a_hi = { hp[(k0 + 4) * LB_], hp[(k0 + 5) * LB_] }; // K = 4..7

  const float* sp = Sws + (((size_t)(l * C_ + c)) * NB_ + kb) * 8;
  v2f b_lo = { sp[k0    ], sp[k0 + 1] };
  v2f b_hi = { sp[k0 + 4], sp[k0 + 5] };

  acc = __builtin_amdgcn_wmma_f32_16x16x4_f32(false, a_lo, false, b_lo,
                                              (short)0, acc, false, false);
  acc = __builtin_amdgcn_wmma_f32_16x16x4_f32(false, a_hi, false, b_hi,
                                              (short)0, acc, false, false);

  float4 o0 = {acc[0], acc[1], acc[2], acc[3]};
  float4 o1 = {acc[4], acc[5], acc[6], acc[7]};
  *(float4*)(yp)     = o0;
  *(float4*)(yp + 4) = o1;
}

// ---------------------------------------------------------------------------
extern "C" void kernel_launch(void* const* d_in, const int* in_sizes, int n_in,
                              void* d_out, int out_size, void* d_ws, size_t ws_size,
                              hipStream_t stream) {
  const float* x  = (const float*)d_in[0];  // [16, 40000]
  const float* bc = (const float*)d_in[1];  // [4, 50, 3]
  const float* ac = (const float*)d_in[2];  // [4, 50, 3]
  float* out = (float*)d_out;               // [16, 50, 40000]

  // Workspace layout (floats): h | E | dstates | Sws   (~3.9 MB total)
  float* ws = (float*)d_ws;
  float* h   = ws;                               // C_*8*LB_ = 320000
  float* E   = h + (size_t)C_ * 8 * LB_;         // C_*8*8   = 3200
  float* dst = E + (size_t)C_ * 8 * 8;           // NCH_*NB_*8 = 320000
  float* Sws = dst + (size_t)NCH_ * NB_ * 8;     // NCH_*NB_*8 = 320000

  // 1) Per-channel initial-state responses + end-state matrices.
  gt_precompute_h<<<dim3((C_ * 8 + 63) / 64), dim3(64), 0, stream>>>(bc, ac, h, E);

  // 2) Zero-state block pass (writes y0 into d_out, end states into dst).
  {
    int total = NCH_ * NB_;
    gt_zero_state<<<dim3((total + 127) / 128), dim3(128), 0, stream>>>(
        x, bc, ac, out, dst);
  }

  // 3) Sequential-over-blocks boundary scan per chain.
  gt_scan<<<dim3((NCH_ + 127) / 128), dim3(128), 0, stream>>>(E, dst, Sws);

  // 4) WMMA fix-up, in place on d_out. Exactly C_*(T_/16) = 125000 tiles,
  //    8 per 256-thread workgroup -> 15625 workgroups, all waves full.
  gt_fixup_wmma<<<dim3(C_ * (T_ / 16) / 8), dim3(256), 0, stream>>>(h, Sws, out);
}